// KnnGraph_33036888441074
// MI455X (gfx1250) — compile-verified
//
#include <hip/hip_runtime.h>

// Problem constants (match reference)
#define BB  8
#define MM  4096
#define KNN 31            // K-1 non-self neighbors per node
#define DD  128
#define NN  (BB * MM)

// d_out layout (floats), outputs concatenated in reference return order
#define OFF_FEAT  0
#define OFF_COORD (NN * DD)                 // 4,194,304
#define OFF_ESRC  (OFF_COORD + NN * 3)      // 4,292,608
#define OFF_EDST  (OFF_ESRC + NN * KNN)     // 5,308,416
#define OFF_DEG   (OFF_EDST + NN * KNN)     // 6,324,224
#define OFF_BATCH (OFF_DEG + NN)            // 6,356,992

#define BIGF 3.0e38f

typedef __attribute__((ext_vector_type(2))) float v2f;
typedef __attribute__((ext_vector_type(4))) float v4f;
typedef __attribute__((ext_vector_type(8))) float v8f;

#if __has_builtin(__builtin_amdgcn_global_load_async_to_lds_b32)
#define USE_ASYNC_LDS 1
typedef __attribute__((address_space(1))) int* g32_ptr;   // global (per clang diag)
typedef __attribute__((address_space(3))) int* l32_ptr;   // LDS
#endif

// ---------------------------------------------------------------------------
// Streaming pass-through of node features (32 MB traffic, HBM-bound).
// ---------------------------------------------------------------------------
__global__ void feat_copy_kernel(const v4f* __restrict__ feat,
                                 v4f* __restrict__ out) {
    int i = blockIdx.x * blockDim.x + threadIdx.x;   // NN*DD/4 items
    v4f v = __builtin_nontemporal_load(&feat[i]);
    __builtin_nontemporal_store(v, &out[i]);
}

// ---------------------------------------------------------------------------
// Closed-form outputs: coord copy, edge_dst = repeat(i,31), degree = 31,
// batch = i / M  (one thread per node)
// ---------------------------------------------------------------------------
__global__ void misc_kernel(const float* __restrict__ coord,
                            float* __restrict__ out) {
    int i = blockIdx.x * blockDim.x + threadIdx.x;   // node index, NN items
    out[OFF_COORD + 3 * i]     = coord[3 * i];
    out[OFF_COORD + 3 * i + 1] = coord[3 * i + 1];
    out[OFF_COORD + 3 * i + 2] = coord[3 * i + 2];
    out[OFF_DEG + i]   = (float)KNN;
    out[OFF_BATCH + i] = (float)(i >> 12);           // i / MM
    size_t eo = (size_t)OFF_EDST + (size_t)i * KNN;
    float fi = (float)i;
#pragma unroll
    for (int j = 0; j < KNN; ++j)
        __builtin_nontemporal_store(fi, &out[eo + j]);
}

// ---------------------------------------------------------------------------
// kNN kernel: 128 threads (4 waves), 64 dst rows per block, 512 blocks.
//  - Gram tiles via V_WMMA_F32_16X16X4_F32 (K=3 padded to 4)
//  - staged value is csq[col] - 2*gram  (rank-equivalent to d^2; rsq is a
//    per-row constant, so ordering and the half-list merge are unchanged)
//  - self-edge removed by patching one staged LDS cell to +BIG
//  - batch coords LDS-resident via async global->LDS (ASYNCcnt-tracked)
//  - per-row selection split across TWO threads, register-resident sorted
//    top-31 each, exact 2-way merge at the end.
// ---------------------------------------------------------------------------
__global__ __launch_bounds__(128) void knn_kernel(const float* __restrict__ coord,
                                                  float* __restrict__ out) {
    __shared__ float sx[MM];            // 16 KB  batch coords, SoA
    __shared__ float sy[MM];            // 16 KB
    __shared__ float sz[MM];            // 16 KB
    __shared__ float gram[64][33];      // 8.4 KB staged (csq - 2*gram), padded

    const int tid        = threadIdx.x;
    const int lane       = tid & 31;
    const int wave       = tid >> 5;
    const int l15        = lane & 15;
    const int halfHi     = (lane >= 16);
    const int blockLocal = blockIdx.x & 63;       // 64 blocks per batch
    const int batch      = blockIdx.x >> 6;
    const int batchBase  = batch * MM;

    // ---- Stage this batch's point cloud into LDS (48 KB), deinterleaved ----
#ifdef USE_ASYNC_LDS
    {
        float* gbase = const_cast<float*>(coord);
        for (int p = tid; p < MM; p += 128) {
            int g = (batchBase + p) * 3;
            __builtin_amdgcn_global_load_async_to_lds_b32(
                (g32_ptr)(gbase + g),     (l32_ptr)&sx[p], 0, 0);
            __builtin_amdgcn_global_load_async_to_lds_b32(
                (g32_ptr)(gbase + g + 1), (l32_ptr)&sy[p], 0, 0);
            __builtin_amdgcn_global_load_async_to_lds_b32(
                (g32_ptr)(gbase + g + 2), (l32_ptr)&sz[p], 0, 0);
        }
    }
#if __has_builtin(__builtin_amdgcn_s_wait_asynccnt)
    __builtin_amdgcn_s_wait_asynccnt(0);
#else
    asm volatile("s_wait_asynccnt 0" ::: "memory");
#endif
#else
    for (int p = tid; p < MM; p += 128) {
        int g = (batchBase + p) * 3;
        sx[p] = coord[g];
        sy[p] = coord[g + 1];
        sz[p] = coord[g + 2];
    }
#endif
    __syncthreads();

    // ---- A fragment (persistent): 16 dst rows of this wave, 16x4 f32 ------
    // ISA layout: lanes 0-15 hold K=0 (VGPR0) / K=1 (VGPR1); lanes 16-31 K=2/K=3.
    const int aRow = blockLocal * 64 + wave * 16 + l15;
    v2f a;
    a.x = halfHi ? sz[aRow] : sx[aRow];
    a.y = halfHi ? 0.0f     : sy[aRow];

    // ---- Selection state: thread t and t+64 share row t (t < 64) ----------
    const int rsel   = tid & 63;                  // row index within block
    const int rhalf  = tid >> 6;                  // 0: cols 0-15, 1: cols 16-31
    const int cb     = rhalf * 16;                // column sub-tile base
    const int selRow = blockLocal * 64 + rsel;    // local row within batch
    float bd[KNN];
    int   bi[KNN];
#pragma unroll
    for (int j = 0; j < KNN; ++j) { bd[j] = BIGF; bi[j] = 0; }

    for (int colBase = 0; colBase < MM; colBase += 32) {
        // Two 16x16 Gram tiles per wave (uniform code -> EXEC all 1s at WMMA)
#pragma unroll
        for (int s = 0; s < 2; ++s) {
            int col = colBase + s * 16 + l15;
            float x = sx[col], y = sy[col], z = sz[col];
            v2f b;
            b.x = halfHi ? z    : x;
            b.y = halfHi ? 0.0f : y;
            float colSq = x * x + y * y + z * z;
            v8f c8 = {};
            c8 = __builtin_amdgcn_wmma_f32_16x16x4_f32(
                    /*neg_a=*/false, a, /*neg_b=*/false, b,
                    /*c_mod=*/(short)0, c8, /*reuse_a=*/false, /*reuse_b=*/false);
            // C/D layout: VGPR r -> row r (lanes 0-15) / row r+8 (lanes 16-31)
            // Stage csq - 2*gram: rank-equivalent to squared distance.
            int rbase = wave * 16 + (halfHi ? 8 : 0);
#pragma unroll
            for (int r = 0; r < 8; ++r)
                gram[rbase + r][s * 16 + l15] = __builtin_fmaf(-2.0f, c8[r], colSq);
        }
        __syncthreads();

        // Patch out the self-edge: only this thread ever reads that cell.
        // (selfCC >> 4) == rhalf  also implies 0 <= selfCC < 32 (arith shift).
        int selfCC = selRow - colBase;
        if ((selfCC >> 4) == rhalf) gram[rsel][selfCC] = BIGF;

        // Selection: all 128 threads; each scans its 16-col half of one row.
        // Hot path per candidate: ds_load + v_cmp + rarely-taken insert.
#pragma unroll 4
        for (int c = 0; c < 16; ++c) {
            int   cc = cb + c;
            float d2 = gram[rsel][cc];
            if (d2 < bd[KNN - 1]) {
                float cd = d2; int ci = colBase + cc;
#pragma unroll
                for (int j = 0; j < KNN; ++j) {       // static idx -> registers
                    bool  lt = cd < bd[j];
                    float od = bd[j]; int oi = bi[j];
                    bd[j] = lt ? cd : od;
                    bi[j] = lt ? ci : oi;
                    cd    = lt ? od : cd;
                    ci    = lt ? oi : ci;
                }
            }
        }
        __syncthreads();
    }

    // ---- Exact merge of the two per-half top-31 lists (reuse LDS) ---------
    float* dA = sx;                     // [64][31] sorted rel-dists, half 0
    int*   iA = (int*)sy;               // [64][31] indices,          half 0
    float* dB = sz;                     // [64][31] sorted rel-dists, half 1
    int*   iB = (int*)gram;             // [64][31] indices, half 1 (2112 >= 1984)
    {
        float* dp = rhalf ? dB : dA;
        int*   ip = rhalf ? iB : iA;
#pragma unroll
        for (int j = 0; j < KNN; ++j) {
            dp[rsel * KNN + j] = bd[j];
            ip[rsel * KNN + j] = bi[j];
        }
    }
    __syncthreads();

    if (tid < 64) {
        size_t eo = (size_t)OFF_ESRC + (size_t)(batchBase + selRow) * KNN;
        int ia = 0, ib = 0;
        for (int j = 0; j < KNN; ++j) {
            float da = (ia < KNN) ? dA[rsel * KNN + ia] : BIGF;
            float db = (ib < KNN) ? dB[rsel * KNN + ib] : BIGF;
            bool ta = da <= db;
            int idx = ta ? iA[rsel * KNN + ia] : iB[rsel * KNN + ib];
            __builtin_nontemporal_store((float)(batchBase + idx), &out[eo + j]);
            if (ta) ++ia; else ++ib;
        }
    }
}

extern "C" void kernel_launch(void* const* d_in, const int* in_sizes, int n_in,
                              void* d_out, int out_size, void* d_ws, size_t ws_size,
                              hipStream_t stream) {
    const float* coord = (const float*)d_in[0];   // node_coord_src  [N,3]
    const float* feat  = (const float*)d_in[1];   // node_feature_src [N,D]
    // d_in[2] (batch_src) implicit: equal-sized sorted batches
    float* out = (float*)d_out;

    feat_copy_kernel<<<(NN * DD / 4) / 256, 256, 0, stream>>>(
        (const v4f*)feat, (v4f*)(out + OFF_FEAT));
    misc_kernel<<<NN / 256, 256, 0, stream>>>(coord, out);
    knn_kernel<<<NN / 64, 128, 0, stream>>>(coord, out);
}